// SetAbstraction_17403207483664
// MI455X (gfx1250) — compile-verified
//
#include <hip/hip_runtime.h>
#include <hip/hip_bf16.h>

#define Bq   8
#define Nn   4096
#define Mm   1024
#define Kn   64
#define Cf   64
#define Hh   128
#define R2   0.04f
#define NEGV (-1e30f)

#define KPAD  68     // layer-1 K padded (67 -> 68, /4 chunks = 17); also W1t row stride
#define H1STR 130    // h1 row stride (bank rotation 2)
#define W2STR 132    // W2t row stride (bank rotation 4)

typedef __attribute__((ext_vector_type(2))) float v2f;
typedef __attribute__((ext_vector_type(8))) float v8f;

// ---------------------------------------------------------------------------
// 1) Farthest point sampling: one block per cloud, 256 threads.
// ---------------------------------------------------------------------------
__global__ __launch_bounds__(256) void fps_kernel(const float* __restrict__ pos,
                                                  float* __restrict__ centers) {
    const int b = blockIdx.x;
    const float* p = pos + (size_t)b * Nn * 3;

    __shared__ float px[Nn], py[Nn], pz[Nn], mind[Nn];
    __shared__ float rv[8];
    __shared__ int   ri[8];
    __shared__ int   s_next;

    const int tid = threadIdx.x;
    for (int i = tid; i < Nn; i += 256) {
        px[i] = p[3 * i + 0];
        py[i] = p[3 * i + 1];
        pz[i] = p[3 * i + 2];
        mind[i] = 3.4e38f;
    }
    __syncthreads();

    int cur = 0;
    if (tid == 0) {
        centers[(size_t)(b * Mm) * 3 + 0] = px[0];
        centers[(size_t)(b * Mm) * 3 + 1] = py[0];
        centers[(size_t)(b * Mm) * 3 + 2] = pz[0];
    }

    const int base = tid * (Nn / 256);   // blocked: thread order == index order
    for (int m = 1; m < Mm; ++m) {
        const float cx = px[cur], cy = py[cur], cz = pz[cur];
        float bestv = -1.0f;
        int   besti = Nn;
        for (int k = 0; k < Nn / 256; ++k) {
            const int i = base + k;
            const float dx = px[i] - cx, dy = py[i] - cy, dz = pz[i] - cz;
            const float d  = dx * dx + dy * dy + dz * dz;
            const float nm = fminf(mind[i], d);
            mind[i] = nm;
            if (nm > bestv) { bestv = nm; besti = i; }   // strictly > keeps lowest idx
        }
        for (int off = 16; off >= 1; off >>= 1) {
            const float ov = __shfl_xor(bestv, off, 32);
            const int   oi = __shfl_xor(besti, off, 32);
            if (ov > bestv || (ov == bestv && oi < besti)) { bestv = ov; besti = oi; }
        }
        const int wave = tid >> 5;
        if ((tid & 31) == 0) { rv[wave] = bestv; ri[wave] = besti; }
        __syncthreads();
        if (tid == 0) {
            float bv = rv[0]; int bi = ri[0];
            for (int w = 1; w < 8; ++w)
                if (rv[w] > bv || (rv[w] == bv && ri[w] < bi)) { bv = rv[w]; bi = ri[w]; }
            s_next = bi;
            centers[(size_t)(b * Mm + m) * 3 + 0] = px[bi];
            centers[(size_t)(b * Mm + m) * 3 + 1] = py[bi];
            centers[(size_t)(b * Mm + m) * 3 + 2] = pz[bi];
        }
        __syncthreads();
        cur = s_next;
    }
}

// ---------------------------------------------------------------------------
// 2) Ball query: one wave per center; ballot+popc compaction keeps the K
//    lowest-index in-ball points. Unfilled slots get -1.
// ---------------------------------------------------------------------------
__global__ __launch_bounds__(256) void ballquery_kernel(const float* __restrict__ pos,
                                                        const float* __restrict__ centers,
                                                        int* __restrict__ nbr) {
    const int gw   = (blockIdx.x * 256 + threadIdx.x) >> 5;   // center id
    const int lane = threadIdx.x & 31;
    if (gw >= Bq * Mm) return;
    const int b = gw / Mm;
    const float* p = pos + (size_t)b * Nn * 3;
    const float cx = centers[(size_t)gw * 3 + 0];
    const float cy = centers[(size_t)gw * 3 + 1];
    const float cz = centers[(size_t)gw * 3 + 2];
    int* out = nbr + (size_t)gw * Kn;

    int cnt = 0;
    for (int j0 = 0; j0 < Nn && cnt < Kn; j0 += 32) {
        const int j = j0 + lane;
        const float dx = p[3 * j + 0] - cx;
        const float dy = p[3 * j + 1] - cy;
        const float dz = p[3 * j + 2] - cz;
        const bool in = (dx * dx + dy * dy + dz * dz) <= R2;
        const unsigned mask = (unsigned)__ballot(in);
        const int slot = cnt + __popc(mask & ((1u << lane) - 1u));
        if (in && slot < Kn) out[slot] = j;
        cnt += __popc(mask);
    }
    const int filled = cnt < Kn ? cnt : Kn;
    for (int s = filled + lane; s < Kn; s += 32) out[s] = -1;
}

// ---------------------------------------------------------------------------
// 3) Fused PointNetConv: gather -> MLP1 (WMMA f32 16x16x4) -> ReLU ->
//    MLP2 (WMMA) -> ReLU -> masked max over K. One block (4 waves) per center.
//    Weights staged TRANSPOSED in LDS so each B fragment is one ds_load_b64;
//    n-tiles processed in pairs for two independent WMMA accumulation chains.
// ---------------------------------------------------------------------------
__global__ __launch_bounds__(128) void mlp_kernel(const float* __restrict__ x,
                                                  const float* __restrict__ pos,
                                                  const float* __restrict__ centers,
                                                  const int* __restrict__ nbr,
                                                  const float* __restrict__ W1,
                                                  const float* __restrict__ b1,
                                                  const float* __restrict__ W2,
                                                  const float* __restrict__ b2,
                                                  float* __restrict__ out) {
    const int cm = blockIdx.x;            // center id in [0, B*M)
    const int b  = cm >> 10;              // / Mm

    __shared__ float feat[Kn][KPAD];      // [64][68]   A of layer 1
    __shared__ float h1[Kn][H1STR];       // [64][130]  A of layer 2
    __shared__ float W1t[Hh][KPAD];       // [128][68]  B of layer 1, transposed (k 67 zeroed)
    __shared__ float W2t[Hh][W2STR];      // [128][132] B of layer 2, transposed
    __shared__ float bias1[Hh], bias2[Hh];
    __shared__ float wmax[4][Hh];
    __shared__ int   validLDS[Kn];

    const int tid = threadIdx.x;

    if (tid < Hh) { bias1[tid] = b1[tid]; bias2[tid] = b2[tid]; }

    // stage W1 transposed: element (r=k, c=n) -> W1t[n][k]; pad k-row 67 with 0
    for (int i = tid; i < (KPAD * Hh) / 4; i += 128) {
        const int e = i * 4, r = e / Hh, c = e % Hh;   // r in [0,68), c multiple of 4
        float4 v = {0.f, 0.f, 0.f, 0.f};
        if (r < Cf + 3) v = *(const float4*)(W1 + (size_t)r * Hh + c);
        W1t[c + 0][r] = v.x;
        W1t[c + 1][r] = v.y;
        W1t[c + 2][r] = v.z;
        W1t[c + 3][r] = v.w;
    }
    // stage W2 transposed
    for (int i = tid; i < (Hh * Hh) / 4; i += 128) {
        const int e = i * 4, r = e / Hh, c = e % Hh;
        const float4 v = *(const float4*)(W2 + (size_t)r * Hh + c);
        W2t[c + 0][r] = v.x;
        W2t[c + 1][r] = v.y;
        W2t[c + 2][r] = v.z;
        W2t[c + 3][r] = v.w;
    }

    // gather feat rows: 2 threads per neighbor row
    const float cx = centers[(size_t)cm * 3 + 0];
    const float cy = centers[(size_t)cm * 3 + 1];
    const float cz = centers[(size_t)cm * 3 + 2];
    {
        const int r = tid >> 1, half = tid & 1;
        const int j  = nbr[(size_t)cm * Kn + r];
        const int jj = j < 0 ? 0 : j;
        const float* xr = x + ((size_t)b * Nn + jj) * Cf + half * 32;
        float* fr = &feat[r][half * 32];
        for (int c = 0; c < 32; c += 4) *(float4*)(fr + c) = *(const float4*)(xr + c);
        if (half) {
            const float* pr = pos + ((size_t)b * Nn + jj) * 3;
            feat[r][64] = pr[0] - cx;
            feat[r][65] = pr[1] - cy;
            feat[r][66] = pr[2] - cz;
            feat[r][67] = 0.f;
        } else {
            validLDS[r] = (j >= 0);
        }
    }
    __syncthreads();

    const int wave = tid >> 5, lane = tid & 31;
    const int lrow = lane & 15, hi = lane >> 4;   // hi=0: lanes 0-15, hi=1: lanes 16-31
    const int R0 = wave * 16;                     // this wave's 16-row M-tile
    const int ksel = hi * 2;                      // lanes 16-31 take K+2/K+3 of each chunk

    // ---- layer 1: [64,68] x [68,128] ----
    v2f a1[KPAD / 4];
    for (int kc = 0; kc < KPAD / 4; ++kc)
        a1[kc] = *(const v2f*)(&feat[R0 + lrow][kc * 4 + ksel]);

    for (int n = 0; n < 8; n += 2) {
        v8f acc0 = {}, acc1 = {};
        const int col0 = n * 16 + lrow, col1 = col0 + 16;
        for (int kc = 0; kc < KPAD / 4; ++kc) {
            const int kb = kc * 4 + ksel;
            const v2f b0 = *(const v2f*)(&W1t[col0][kb]);
            const v2f b1v = *(const v2f*)(&W1t[col1][kb]);
            acc0 = __builtin_amdgcn_wmma_f32_16x16x4_f32(false, a1[kc], false, b0,
                                                         (short)0, acc0, false, false);
            acc1 = __builtin_amdgcn_wmma_f32_16x16x4_f32(false, a1[kc], false, b1v,
                                                         (short)0, acc1, false, false);
        }
        const float bv0 = bias1[col0], bv1 = bias1[col1];
        for (int v = 0; v < 8; ++v) {                 // D: row = v + 8*hi, col
            h1[R0 + v + 8 * hi][col0] = fmaxf(acc0[v] + bv0, 0.f);
            h1[R0 + v + 8 * hi][col1] = fmaxf(acc1[v] + bv1, 0.f);
        }
    }
    __syncthreads();

    // ---- layer 2: [64,128] x [128,128] + fused masked max over rows ----
    v2f a2[Hh / 4];
    for (int kc = 0; kc < Hh / 4; ++kc)
        a2[kc] = *(const v2f*)(&h1[R0 + lrow][kc * 4 + ksel]);

    for (int n = 0; n < 8; n += 2) {
        v8f acc0 = {}, acc1 = {};
        const int col0 = n * 16 + lrow, col1 = col0 + 16;
        for (int kc = 0; kc < Hh / 4; ++kc) {
            const int kb = kc * 4 + ksel;
            const v2f b0 = *(const v2f*)(&W2t[col0][kb]);
            const v2f b1v = *(const v2f*)(&W2t[col1][kb]);
            acc0 = __builtin_amdgcn_wmma_f32_16x16x4_f32(false, a2[kc], false, b0,
                                                         (short)0, acc0, false, false);
            acc1 = __builtin_amdgcn_wmma_f32_16x16x4_f32(false, a2[kc], false, b1v,
                                                         (short)0, acc1, false, false);
        }
        const float bv0 = bias2[col0], bv1 = bias2[col1];
        float mv0 = NEGV, mv1 = NEGV;
        for (int v = 0; v < 8; ++v) {
            const int ok = validLDS[R0 + v + 8 * hi];
            float h0 = fmaxf(acc0[v] + bv0, 0.f);
            float h1v = fmaxf(acc1[v] + bv1, 0.f);
            mv0 = fmaxf(mv0, ok ? h0 : NEGV);
            mv1 = fmaxf(mv1, ok ? h1v : NEGV);
        }
        mv0 = fmaxf(mv0, __shfl_xor(mv0, 16, 32));    // combine row halves
        mv1 = fmaxf(mv1, __shfl_xor(mv1, 16, 32));
        if (hi == 0) { wmax[wave][col0] = mv0; wmax[wave][col1] = mv1; }
    }
    __syncthreads();

    if (tid < Hh) {
        const float mv = fmaxf(fmaxf(wmax[0][tid], wmax[1][tid]),
                               fmaxf(wmax[2][tid], wmax[3][tid]));
        out[(size_t)cm * Hh + tid] = mv;
    }
}

// ---------------------------------------------------------------------------
extern "C" void kernel_launch(void* const* d_in, const int* in_sizes, int n_in,
                              void* d_out, int out_size, void* d_ws, size_t ws_size,
                              hipStream_t stream) {
    const float* x   = (const float*)d_in[0];
    const float* pos = (const float*)d_in[1];
    const float* W1  = (const float*)d_in[2];
    const float* b1  = (const float*)d_in[3];
    const float* W2  = (const float*)d_in[4];
    const float* b2  = (const float*)d_in[5];

    float* out     = (float*)d_out;                       // [B,M,H]
    float* centers = out + (size_t)Bq * Mm * Hh;          // [B,M,3]
    int*   nbr     = (int*)d_ws;                          // [B,M,K]

    fps_kernel<<<Bq, 256, 0, stream>>>(pos, centers);
    ballquery_kernel<<<(Bq * Mm) / 8, 256, 0, stream>>>(pos, centers, nbr);
    mlp_kernel<<<Bq * Mm, 128, 0, stream>>>(x, pos, centers, nbr, W1, b1, W2, b2, out);
}